// LRGS_MC_Mamba_Block_32134945308973
// MI455X (gfx1250) — compile-verified
//
#include <hip/hip_runtime.h>
#include <hip/hip_bf16.h>

#define BATCH 8
#define CC    64
#define HW    4096
#define DI    128

typedef float v2f __attribute__((ext_vector_type(2)));
typedef float v8f __attribute__((ext_vector_type(8)));

__device__ __forceinline__ float lrelu01(float v){ return v > 0.f ? v : 0.01f * v; }
__device__ __forceinline__ float siluf(float v){ return v / (1.f + __expf(-v)); }

// ---------------------------------------------------------------------------
// LRGS: per (b,c) plane -> dw3/dw5/dw7 conv + InstanceNorm + LeakyReLU,
// written to feats (B, 3C, H, W) in concat order [dw3 | dw5 | dw7].
// ---------------------------------------------------------------------------
__global__ void k_lrgs(const float* __restrict__ x,
                       const float* __restrict__ w3, const float* __restrict__ b3,
                       const float* __restrict__ w5, const float* __restrict__ b5,
                       const float* __restrict__ w7, const float* __restrict__ b7,
                       float* __restrict__ feats)
{
  __shared__ float t[70 * 71];           // 64x64 plane + halo 3, padded row
  __shared__ float sw3[9], sw5[25], sw7[49];
  __shared__ float rs[256], rq[256];
  __shared__ float s_mu, s_rstd;
  const int blk = blockIdx.x, tid = threadIdx.x;
  const int b = blk / CC, c = blk % CC;
  const float* xp = x + (size_t)(b * CC + c) * HW;

  if (tid < 9)  sw3[tid] = w3[c * 9 + tid];
  if (tid < 25) sw5[tid] = w5[c * 25 + tid];
  if (tid < 49) sw7[tid] = w7[c * 49 + tid];
  for (int e = tid; e < 70 * 70; e += 256) {
    int r = e / 70, q = e % 70;
    int gh = r - 3, gw = q - 3;
    float v = 0.f;
    if (gh >= 0 && gh < 64 && gw >= 0 && gw < 64) v = xp[gh * 64 + gw];
    t[r * 71 + q] = v;
  }
  __syncthreads();

  float v3[16], v5[16], v7[16];
  const float bb3 = b3[c], bb5 = b5[c], bb7 = b7[c];
  for (int i = 0; i < 16; i++) {
    int p = tid + i * 256;
    int h = p >> 6, w = p & 63;
    float a3 = bb3, a5 = bb5, a7 = bb7;
    for (int kh = 0; kh < 7; kh++)
      for (int kw = 0; kw < 7; kw++)
        a7 += t[(h + kh) * 71 + (w + kw)] * sw7[kh * 7 + kw];
    for (int kh = 0; kh < 5; kh++)
      for (int kw = 0; kw < 5; kw++)
        a5 += t[(h + 1 + kh) * 71 + (w + 1 + kw)] * sw5[kh * 5 + kw];
    for (int kh = 0; kh < 3; kh++)
      for (int kw = 0; kw < 3; kw++)
        a3 += t[(h + 2 + kh) * 71 + (w + 2 + kw)] * sw3[kh * 3 + kw];
    v3[i] = a3; v5[i] = a5; v7[i] = a7;
  }

#define NORM_STORE(VARR, SIDX)                                                    \
  {                                                                               \
    float s = 0.f, q = 0.f;                                                       \
    for (int i = 0; i < 16; i++) { s += VARR[i]; q += VARR[i] * VARR[i]; }        \
    rs[tid] = s; rq[tid] = q; __syncthreads();                                    \
    for (int o = 128; o > 0; o >>= 1) {                                           \
      if (tid < o) { rs[tid] += rs[tid + o]; rq[tid] += rq[tid + o]; }            \
      __syncthreads();                                                            \
    }                                                                             \
    if (tid == 0) {                                                               \
      float mu = rs[0] * (1.f / 4096.f);                                          \
      float var = rq[0] * (1.f / 4096.f) - mu * mu;                               \
      s_mu = mu; s_rstd = rsqrtf(var + 1e-5f);                                    \
    }                                                                             \
    __syncthreads();                                                              \
    { float mu = s_mu, rstd = s_rstd;                                             \
      float* op = feats + (size_t)(b * (3 * CC) + SIDX * CC + c) * HW;            \
      for (int i = 0; i < 16; i++) {                                              \
        int p = tid + i * 256;                                                    \
        op[p] = lrelu01((VARR[i] - mu) * rstd);                                   \
      } }                                                                         \
    __syncthreads();                                                              \
  }

  NORM_STORE(v3, 0)
  NORM_STORE(v5, 1)
  NORM_STORE(v7, 2)
#undef NORM_STORE
}

// ---------------------------------------------------------------------------
// InstanceNorm (+ optional LeakyReLU) + residual add, per (b,c) plane.
// ---------------------------------------------------------------------------
__global__ void k_inorm_res(const float* __restrict__ src, const float* __restrict__ resid,
                            float* __restrict__ out, int apply_lrelu)
{
  __shared__ float rs[256], rq[256];
  __shared__ float s_mu, s_rstd;
  const int blk = blockIdx.x, tid = threadIdx.x;
  const float* sp = src + (size_t)blk * HW;
  const float* rp = resid + (size_t)blk * HW;
  float* op = out + (size_t)blk * HW;
  float v[16];
  float s = 0.f, q = 0.f;
  for (int i = 0; i < 16; i++) {
    float t = sp[tid + i * 256];
    v[i] = t; s += t; q += t * t;
  }
  rs[tid] = s; rq[tid] = q; __syncthreads();
  for (int o = 128; o > 0; o >>= 1) {
    if (tid < o) { rs[tid] += rs[tid + o]; rq[tid] += rq[tid + o]; }
    __syncthreads();
  }
  if (tid == 0) {
    float mu = rs[0] * (1.f / 4096.f);
    float var = rq[0] * (1.f / 4096.f) - mu * mu;
    s_mu = mu; s_rstd = rsqrtf(var + 1e-5f);
  }
  __syncthreads();
  for (int i = 0; i < 16; i++) {
    float t = (v[i] - s_mu) * s_rstd;
    if (apply_lrelu) t = lrelu01(t);
    op[tid + i * 256] = t + rp[tid + i * 256];
  }
}

// ---------------------------------------------------------------------------
// Generic fp32 WMMA GEMM:  D[b][m][n] = sum_k Aw[m][k] * Bact[b][k][n] + bias[m]
// N fixed at HW=4096. One 16x16 tile per wave, K stepped by 4 through
// V_WMMA_F32_16X16X4_F32 (full fp32 precision, matches reference).
//
// The 16xK A tile is staged into LDS once per workgroup (single guarded
// cooperative loop) so the hot loop has NO divergent exec-masked loads:
// A comes from uniform ds_load, B from two coalesced global_load_b32 with a
// global_prefetch_b8 covering the next K step (stride 16KB through L2).
//
// A layout (ISA 7.12.2): lane%16 = M row, lane/16 selects K pair {0,1}/{2,3};
// B symmetric with lane%16 = N column. C/D: vgpr r -> M = r + 8*(lane/16).
// GUARD=true only for the M=36 (x_proj) call; all other GEMMs get
// straight-line stores.
// ---------------------------------------------------------------------------
#define KMAX 192
template <bool GUARD>
__global__ void k_gemm_wmma(const float* __restrict__ Aw, const float* __restrict__ Bact,
                            const float* __restrict__ bias, float* __restrict__ Dout,
                            int Mi, int Mtiles, int K)
{
  __shared__ float sA[16 * KMAX];                 // 12 KB max
  const int NGRP = HW / 64;                       // 64 groups of 4 n-tiles
  const int wg = blockIdx.x;
  const int b   = wg / (Mtiles * NGRP);
  const int rem = wg % (Mtiles * NGRP);
  const int mt  = rem / NGRP;
  const int ng  = rem % NGRP;
  const int tid  = threadIdx.x;
  const int wave = tid >> 5;
  const int lane = tid & 31;
  const int lo = lane & 15, hi = lane >> 4;
  const int nt = ng * 4 + wave;
  const int col = nt * 16 + lo;

  // Stage A tile (rows mt*16 .. mt*16+15) into LDS, zero-padding rows >= Mi.
  for (int e = tid; e < 16 * K; e += 128) {
    int row = e / K, k = e - row * K;
    int m = mt * 16 + row;
    sA[e] = (m < Mi) ? Aw[m * K + k] : 0.f;
  }
  __syncthreads();

  const float* Brow = Bact + (size_t)b * K * HW + col;
  const float* aRow = sA + lo * K + 2 * hi;       // this lane's K pair base

  v8f acc = {0.f, 0.f, 0.f, 0.f, 0.f, 0.f, 0.f, 0.f};
#pragma unroll 4
  for (int kk = 0; kk < K; kk += 4) {
    v2f a, bv;
    a.x = aRow[kk];
    a.y = aRow[kk + 1];
    const float* bp = Brow + (size_t)(kk + 2 * hi) * HW;
    __builtin_prefetch(bp + 4 * HW, 0, 3);        // global_prefetch_b8, next K step
    bv.x = bp[0];
    bv.y = bp[HW];
    acc = __builtin_amdgcn_wmma_f32_16x16x4_f32(false, a, false, bv,
                                                (short)0, acc, false, false);
  }
  for (int r = 0; r < 8; r++) {
    const int mo = mt * 16 + r + 8 * hi;
    if (!GUARD || mo < Mi) {
      float v = acc[r] + (bias ? bias[mo] : 0.f);
      Dout[((size_t)b * Mi + mo) * HW + col] = v;
    }
  }
}

// ---------------------------------------------------------------------------
// Depthwise 3x3 conv + bias + SiLU on xm (= xz rows 0..127 per batch) -> u.
// ---------------------------------------------------------------------------
__global__ void k_dwsilu(const float* __restrict__ xz, const float* __restrict__ cw,
                         const float* __restrict__ cb, float* __restrict__ u)
{
  __shared__ float t[66 * 67];
  __shared__ float sw[9];
  const int blk = blockIdx.x, tid = threadIdx.x;
  const int b = blk / DI, d = blk % DI;
  const float* sp = xz + (size_t)(b * (2 * DI) + d) * HW;
  if (tid < 9) sw[tid] = cw[d * 9 + tid];
  for (int e = tid; e < 66 * 66; e += 256) {
    int r = e / 66, q = e % 66;
    int gh = r - 1, gw = q - 1;
    float v = 0.f;
    if (gh >= 0 && gh < 64 && gw >= 0 && gw < 64) v = sp[gh * 64 + gw];
    t[r * 67 + q] = v;
  }
  __syncthreads();
  const float bi = cb[d];
  float* op = u + (size_t)(b * DI + d) * HW;
  for (int i = 0; i < 16; i++) {
    int p = tid + i * 256;
    int h = p >> 6, w = p & 63;
    float a = bi;
    for (int kh = 0; kh < 3; kh++)
      for (int kw = 0; kw < 3; kw++)
        a += t[(h + kh) * 67 + (w + kw)] * sw[kh * 3 + kw];
    op[p] = siluf(a);
  }
}

// ---------------------------------------------------------------------------
// delta[b,d,p] = softplus( dt_raw[b,:,p] . dt_proj_w[d,:] + 2*dt_proj_b[d] )
// (reference applies dt_proj_b twice: Linear bias + delta_bias in the scan)
// ---------------------------------------------------------------------------
__global__ void k_delta(const float* __restrict__ xp, const float* __restrict__ dtw,
                        const float* __restrict__ dtb, float* __restrict__ delta)
{
  const int idx = blockIdx.x * 256 + threadIdx.x;  // B*DI*HW
  const int p = idx & (HW - 1);
  const int d = (idx >> 12) & (DI - 1);
  const int b = idx >> 19;
  const float* base = xp + (size_t)b * 36 * HW + p;
  float v = 2.f * dtb[d];
  for (int r = 0; r < 4; r++) v += dtw[d * 4 + r] * base[r * HW];
  delta[idx] = (v > 20.f) ? v : log1pf(__expf(v));
}

// ---------------------------------------------------------------------------
// Selective scan. 64 WGs x 256 threads; thread = (d_local in 0..15, n in 0..15).
// LDS-staged 128-step chunks; h = exp(delta*A)*h + delta*u*B; y = sum_n h*C
// reduced with 16-lane shfl_xor; + u*Dp skip connection.
// ---------------------------------------------------------------------------
#define SROW 129
__global__ void k_scan(const float* __restrict__ delta, const float* __restrict__ u,
                       const float* __restrict__ xp, const float* __restrict__ A_log,
                       const float* __restrict__ Dp, float* __restrict__ y)
{
  __shared__ float sd[16 * SROW], su[16 * SROW], sB[16 * SROW], sC[16 * SROW];
  const int blk = blockIdx.x, tid = threadIdx.x;
  const int b = blk >> 3, dblk = blk & 7;
  const int dl = tid >> 4, n = tid & 15;
  const int d = dblk * 16 + dl;
  const float Aval = -__expf(A_log[d * 16 + n]);
  const float Dval = Dp[d];
  float h = 0.f;
  const float* dbase = delta + (size_t)b * DI * HW;
  const float* ubase = u + (size_t)b * DI * HW;
  const float* Bbase = xp + (size_t)(b * 36 + 4) * HW;    // Bm rows 4..19
  const float* Cbase = xp + (size_t)(b * 36 + 20) * HW;   // Cm rows 20..35
  float* ybase = y + (size_t)(b * DI + d) * HW;

  for (int c0 = 0; c0 < HW; c0 += 128) {
    for (int j = 0; j < 8; j++) {
      int e = tid + j * 256;
      int r = e >> 7, l = e & 127;
      sd[r * SROW + l] = dbase[(dblk * 16 + r) * HW + c0 + l];
      su[r * SROW + l] = ubase[(dblk * 16 + r) * HW + c0 + l];
      sB[r * SROW + l] = Bbase[r * HW + c0 + l];
      sC[r * SROW + l] = Cbase[r * HW + c0 + l];
    }
    __syncthreads();
    for (int l = 0; l < 128; l++) {
      float ds = sd[dl * SROW + l];
      float uu = su[dl * SROW + l];
      float dA = __expf(ds * Aval);
      h = dA * h + ds * uu * sB[n * SROW + l];
      float part = h * sC[n * SROW + l];
      part += __shfl_xor(part, 8, 32);
      part += __shfl_xor(part, 4, 32);
      part += __shfl_xor(part, 2, 32);
      part += __shfl_xor(part, 1, 32);
      if (n == 0) ybase[c0 + l] = part + uu * Dval;
    }
    __syncthreads();
  }
}

// ---------------------------------------------------------------------------
// LayerNorm over d (128) at each (b,p), * silu(z). In-place on y (one thread
// owns all d at its p, so no cross-thread hazard).
// ---------------------------------------------------------------------------
__global__ void k_ynorm(float* __restrict__ y, const float* __restrict__ xz,
                        const float* __restrict__ nw, const float* __restrict__ nb)
{
  const int idx = blockIdx.x * 256 + threadIdx.x;  // B*HW
  const int p = idx & (HW - 1);
  const int b = idx >> 12;
  float* ybase = y + (size_t)b * DI * HW + p;
  const float* zbase = xz + (size_t)(b * 2 * DI + DI) * HW + p;
  float s = 0.f, q = 0.f;
  for (int d2 = 0; d2 < DI; d2++) {
    float v = ybase[d2 * HW];
    s += v; q += v * v;
  }
  const float mu = s * (1.f / DI);
  const float var = q * (1.f / DI) - mu * mu;
  const float rstd = rsqrtf(var + 1e-5f);
  for (int d2 = 0; d2 < DI; d2++) {
    float v = ybase[d2 * HW];
    float zz = zbase[d2 * HW];
    ybase[d2 * HW] = ((v - mu) * rstd * nw[d2] + nb[d2]) * siluf(zz);
  }
}

// ---------------------------------------------------------------------------
extern "C" void kernel_launch(void* const* d_in, const int* in_sizes, int n_in,
                              void* d_out, int out_size, void* d_ws, size_t ws_size,
                              hipStream_t stream) {
  const float* x         = (const float*)d_in[0];
  const float* dw3_w     = (const float*)d_in[1];
  const float* dw3_b     = (const float*)d_in[2];
  const float* dw5_w     = (const float*)d_in[3];
  const float* dw5_b     = (const float*)d_in[4];
  const float* dw7_w     = (const float*)d_in[5];
  const float* dw7_b     = (const float*)d_in[6];
  const float* fuse_w    = (const float*)d_in[7];
  const float* fuse_b    = (const float*)d_in[8];
  const float* in_proj_w = (const float*)d_in[9];
  const float* conv2d_w  = (const float*)d_in[10];
  const float* conv2d_b  = (const float*)d_in[11];
  const float* x_proj_w  = (const float*)d_in[12];
  const float* dt_proj_w = (const float*)d_in[13];
  const float* dt_proj_b = (const float*)d_in[14];
  const float* A_log     = (const float*)d_in[15];
  const float* Dp        = (const float*)d_in[16];
  const float* out_norm_w= (const float*)d_in[17];
  const float* out_norm_b= (const float*)d_in[18];
  const float* out_proj_w= (const float*)d_in[19];
  const float* cf_w      = (const float*)d_in[20];
  const float* cf_b      = (const float*)d_in[21];

  float* ws = (float*)d_ws;
  // Workspace layout (floats). feats region is reused for xp+delta after the
  // fuse GEMM; xf is reused for y2 after x1 is formed; u is reused for cfo
  // after the scan. Total = 27,262,976 floats (~104 MiB).
  float* feats = ws;                     // B*192*HW = 6,291,456
  float* xp    = ws;                     // alias feats: B*36*HW = 1,179,648
  float* delta = ws + 1179648;           // B*128*HW = 4,194,304 (inside feats)
  float* xf    = ws + 6291456;           // B*64*HW  = 2,097,152
  float* y2    = xf;                     // alias xf
  float* x1    = ws + 8388608;           // B*64*HW  = 2,097,152
  float* xz    = ws + 10485760;          // B*256*HW = 8,388,608
  float* u     = ws + 18874368;          // B*128*HW = 4,194,304
  float* cfo   = u;                      // alias u
  float* y     = ws + 23068672;          // B*128*HW = 4,194,304

  // 1) LRGS multi-scale depthwise + inorm + lrelu -> feats
  k_lrgs<<<BATCH * CC, 256, 0, stream>>>(x, dw3_w, dw3_b, dw5_w, dw5_b, dw7_w, dw7_b, feats);
  // 2) fuse 1x1 conv GEMM (M=64, K=192) -> xf
  k_gemm_wmma<false><<<BATCH * 4 * 64, 128, 0, stream>>>(fuse_w, feats, fuse_b, xf, 64, 4, 192);
  // 3) inorm + lrelu + residual -> x1
  k_inorm_res<<<BATCH * CC, 256, 0, stream>>>(xf, x, x1, 1);
  // 4) in_proj GEMM (M=256, K=64) -> xz  (rows 0..127 = xm, 128..255 = z)
  k_gemm_wmma<false><<<BATCH * 16 * 64, 128, 0, stream>>>(in_proj_w, x1, nullptr, xz, 256, 16, 64);
  // 5) depthwise 3x3 + SiLU -> u
  k_dwsilu<<<BATCH * DI, 256, 0, stream>>>(xz, conv2d_w, conv2d_b, u);
  // 6) x_proj GEMM (M=36, K=128) -> xp (dt rows 0..3, B rows 4..19, C rows 20..35)
  k_gemm_wmma<true><<<BATCH * 3 * 64, 128, 0, stream>>>(x_proj_w, u, nullptr, xp, 36, 3, 128);
  // 7) dt_proj (K=4) + double bias + softplus -> delta
  k_delta<<<(BATCH * DI * HW) / 256, 256, 0, stream>>>(xp, dt_proj_w, dt_proj_b, delta);
  // 8) selective scan -> y
  k_scan<<<BATCH * 8, 256, 0, stream>>>(delta, u, xp, A_log, Dp, y);
  // 9) layernorm * silu(z), in place on y
  k_ynorm<<<(BATCH * HW) / 256, 256, 0, stream>>>(y, xz, out_norm_w, out_norm_b);
  // 10) out_proj GEMM (M=64, K=128) -> y2
  k_gemm_wmma<false><<<BATCH * 4 * 64, 128, 0, stream>>>(out_proj_w, y, nullptr, y2, 64, 4, 128);
  // 11) cf 1x1 conv GEMM (M=64, K=64) -> cfo
  k_gemm_wmma<false><<<BATCH * 4 * 64, 128, 0, stream>>>(cf_w, y2, cf_b, cfo, 64, 4, 64);
  // 12) inorm + residual x1 -> out
  k_inorm_res<<<BATCH * CC, 256, 0, stream>>>(cfo, x1, (float*)d_out, 0);
}